// AttentionPruneKV_77996606096031
// MI455X (gfx1250) — compile-verified
//
#include <hip/hip_runtime.h>

typedef __attribute__((ext_vector_type(2))) float v2f;
typedef __attribute__((ext_vector_type(8))) float v8f;
typedef __attribute__((ext_vector_type(4))) unsigned int u32x4;
typedef __attribute__((ext_vector_type(8))) int i32x8;
typedef __attribute__((ext_vector_type(4))) int i32x4;

#define WMMA_F32X4(a, b, c) \
  __builtin_amdgcn_wmma_f32_16x16x4_f32(false, (a), false, (b), (short)0, (c), false, false)

#define B_ 4
#define N_ 1024
#define DIM_ 512
#define H_ 8
#define DH_ 64

__device__ __forceinline__ float sigmoidf_(float x) { return 1.0f / (1.0f + __expf(-x)); }

__device__ __forceinline__ float rmax16(float v) {
#pragma unroll
  for (int m = 8; m >= 1; m >>= 1) v = fmaxf(v, __shfl_xor(v, m, 16));
  return v;
}
__device__ __forceinline__ float rmin16(float v) {
#pragma unroll
  for (int m = 8; m >= 1; m >>= 1) v = fminf(v, __shfl_xor(v, m, 16));
  return v;
}
__device__ __forceinline__ float rsum16(float v) {
#pragma unroll
  for (int m = 8; m >= 1; m >>= 1) v += __shfl_xor(v, m, 16);
  return v;
}

// ---- Tensor Data Mover: 2D f32 tile Global->LDS per CDNA5 D# layout (ISA 08 §8) ----
// group0: [1:0]=count=1, [63:32]=lds_addr, [120:64]=global_addr, [127:126]=type(2)
// group1: [17:16]=data_size(2=4B), [79:48]=tensor_dim0, [111:80]=tensor_dim1,
//         [127:112]=tile_dim0, [143:128]=tile_dim1, [207:160]=tensor_dim0_stride
__device__ __forceinline__ void tdm_load_2d_f32(const void* gaddr, unsigned lds_off,
                                                unsigned tile_w, unsigned tile_h,
                                                unsigned long long tensor_w,
                                                unsigned long long tensor_h,
                                                unsigned long long row_stride) {
  union { unsigned long long q[2]; u32x4 v; } g0;
  union { unsigned long long q[4]; i32x8 v; } g1;
  const unsigned long long ga = (unsigned long long)(size_t)gaddr;
  g0.q[0] = 1ull | ((unsigned long long)lds_off << 32);
  g0.q[1] = (ga & 0x1FFFFFFFFFFFFFFull) | (2ull << 62);
  g1.q[0] = (2ull << 16) | ((tensor_w & 0xFFFFull) << 48);
  g1.q[1] = ((tensor_w >> 16) & 0xFFFFull) | ((tensor_h & 0xFFFFFFFFull) << 16) |
            ((unsigned long long)(tile_w & 0xFFFFu) << 48);
  g1.q[2] = (unsigned long long)(tile_h & 0xFFFFu) | ((row_stride & 0xFFFFFFFFull) << 32);
  g1.q[3] = (row_stride >> 32) & 0xFFFFull;
  const i32x4 z4 = {0, 0, 0, 0};
#if __clang_major__ >= 23
  const i32x8 z8 = {0, 0, 0, 0, 0, 0, 0, 0};
  __builtin_amdgcn_tensor_load_to_lds(g0.v, g1.v, z4, z4, z8, 0);
#else
  __builtin_amdgcn_tensor_load_to_lds(g0.v, g1.v, z4, z4, 0);
#endif
}

// ---------------- Kernel 1: QKV projection ----------------
// x [4096, 512] @ w_qkv [512, 1536] -> q/k/v each [B,H,N,DH]
__global__ __launch_bounds__(256) void k_qkv(const float* __restrict__ x,
                                             const float* __restrict__ w,
                                             float* __restrict__ q,
                                             float* __restrict__ k,
                                             float* __restrict__ v) {
  __shared__ float smA[16 * 512];  // 32 KB A tile, DMA'd by the Tensor Data Mover
  const int tid = threadIdx.x;
  const int wv = tid >> 5, lane = tid & 31, kh = lane >> 4, ln = lane & 15;
  const int m0 = blockIdx.x * 16;

  if (wv == 0) {
    tdm_load_2d_f32(x + (size_t)m0 * 512, (unsigned)(size_t)(void*)smA,
                    /*tile_w=*/512, /*tile_h=*/16,
                    /*tensor_w=*/512, /*tensor_h=*/4096, /*row_stride=*/512);
    __builtin_amdgcn_s_wait_tensorcnt(0);
  }
  __syncthreads();

  const int ntile = blockIdx.y * 8 + wv;
  const int col0 = ntile * 16;
  v8f acc = {0.f, 0.f, 0.f, 0.f, 0.f, 0.f, 0.f, 0.f};
#pragma unroll 4
  for (int kk = 0; kk < 128; kk++) {
    const int k0 = kk * 4 + kh * 2;
    v2f a = *(const v2f*)&smA[ln * 512 + k0];
    v2f bv;
    bv.x = w[(size_t)k0 * 1536 + col0 + ln];
    bv.y = w[(size_t)(k0 + 1) * 1536 + col0 + ln];
    acc = WMMA_F32X4(a, bv, acc);
  }
  // scatter to q/k/v in [B,H,N,DH] layout (16-col tile never crosses part/head boundary)
  const int bb = m0 >> 10, nloc = m0 & 1023;
  const int col = col0 + ln, part = col >> 9, rem = col & 511, hh = rem >> 6, dd = rem & 63;
  float* dst = (part == 0) ? q : ((part == 1) ? k : v);
  float* p = dst + ((size_t)(bb * H_ + hh) * N_ + nloc) * DH_ + dd;
#pragma unroll
  for (int r = 0; r < 8; r++) {
    const int mr = r + 8 * kh;
    p[(size_t)mr * DH_] = acc[r];
  }
}

// ---------------- Kernel 2: fused attention (per (b,h) x 16-query block) ----------------
__global__ __launch_bounds__(256) void k_attn(const float* __restrict__ qws,
                                              const float* __restrict__ kws,
                                              const float* __restrict__ vws,
                                              const float* __restrict__ prob,
                                              const float* __restrict__ rpb_table,
                                              const float* __restrict__ headsita,
                                              const float* __restrict__ w_th,
                                              const int* __restrict__ rpi,
                                              const float* __restrict__ dis,
                                              float* __restrict__ out_heads,
                                              float* __restrict__ out_sm0) {
  extern __shared__ float smem[];
  float* sm_q = smem;                     // 16 x 64
  float* sm_d0 = smem + 1024;             // 16 x 1024 (dots0 -> exp -> softmax0)
  float* sm_dd = smem + 1024 + 16 * 1024; // 16 x 1024 (dots  -> exp -> attn_p)

  const int tid = threadIdx.x, wv = tid >> 5, lane = tid & 31, kh = lane >> 4, ln = lane & 15;
  const int bh = blockIdx.x >> 6, rowblk = blockIdx.x & 63;
  const int bb = bh >> 3, hh = bh & 7;
  const int m0 = rowblk * 16;
  const float* qb = qws + (size_t)bh * N_ * DH_;
  const float* kb = kws + (size_t)bh * N_ * DH_;
  const float* vb = vws + (size_t)bh * N_ * DH_;

  // Q tile (16x64, contiguous 4 KB) via Tensor Data Mover
  if (wv == 0) {
    tdm_load_2d_f32(qb + (size_t)m0 * DH_, (unsigned)(size_t)(void*)sm_q,
                    /*tile_w=*/1024, /*tile_h=*/1,
                    /*tensor_w=*/1024, /*tensor_h=*/1, /*row_stride=*/1024);
    __builtin_amdgcn_s_wait_tensorcnt(0);
  }
  __syncthreads();

  const float hs = headsita[hh];
  const float factor = 1.0f / (2.0f * hs * hs + 1e-6f);

  // Phase B: S = q @ k^T (scale), + RPE biases; 8 waves x 8 ntiles
  for (int nt = wv; nt < 64; nt += 8) {
    if (nt + 8 < 64) __builtin_prefetch(&kb[(size_t)((nt + 8) * 16 + ln) * DH_], 0, 1);
    v8f acc = {0.f, 0.f, 0.f, 0.f, 0.f, 0.f, 0.f, 0.f};
#pragma unroll
    for (int kk = 0; kk < 16; kk++) {
      const int k0 = kk * 4 + kh * 2;
      v2f a = *(const v2f*)&sm_q[ln * 64 + k0];
      v2f bv = *(const v2f*)&kb[(size_t)(nt * 16 + ln) * DH_ + k0];  // contiguous per lane
      acc = WMMA_F32X4(a, bv, acc);
    }
#pragma unroll
    for (int r = 0; r < 8; r++) {
      const int mr = r + 8 * kh, c = nt * 16 + ln, ig = m0 + mr;
      const float d0v = acc[r] * 0.125f;  // DH^-0.5
      const int idx = rpi[ig * N_ + c];
      const float bias = rpb_table[idx * H_ + hh] + 0.01f * __expf(-factor * dis[ig * N_ + c]);
      sm_d0[mr * 1024 + c] = d0v;
      sm_dd[mr * 1024 + c] = d0v + bias;
    }
  }
  __syncthreads();

  // Phase C: per-row softmax stats + threshold prune (wave owns 2 rows, 16 lanes each)
  const int row = wv * 2 + kh;
  float* d0r = sm_d0 + row * 1024;
  float* ddr = sm_dd + row * 1024;
  const int c0 = ln * 64;

  float mx0 = -1e30f, mxd = -1e30f, mnd = 1e30f;
  for (int j = 0; j < 64; j++) {
    const float a = d0r[c0 + j], d = ddr[c0 + j];
    mx0 = fmaxf(mx0, a);
    mxd = fmaxf(mxd, d);
    mnd = fminf(mnd, d);
  }
  mx0 = rmax16(mx0); mxd = rmax16(mxd); mnd = rmin16(mnd);

  float s0 = 0.f, sd = 0.f;
  for (int j = 0; j < 64; j++) {
    const float e0 = __expf(d0r[c0 + j] - mx0); d0r[c0 + j] = e0; s0 += e0;
    const float ed = __expf(ddr[c0 + j] - mxd); ddr[c0 + j] = ed; sd += ed;
  }
  s0 = rsum16(s0); sd = rsum16(sd);

  float qd = 0.f;
#pragma unroll
  for (int j = 0; j < 4; j++) qd += sm_q[row * 64 + ln * 4 + j] * w_th[ln * 4 + j];
  qd = rsum16(qd);
  const float tmix = sigmoidf_(qd) * 0.11920292202211755f;  // sigmoid(-2.0)
  const float amax = 1.0f / sd;                  // exp(max-max)/sum
  const float amin = __expf(mnd - mxd) / sd;     // exp(min-max)/sum
  const float cmpE = (amin + tmix * (amax - amin)) * sd;  // compare in exp-space

  const int ig = m0 + row;
  const float rowmask = (prob[bb * N_ + ig] < 0.9f) ? 1.0f : 0.0f;

  float denoE = 0.f;
  for (int j = 0; j < 64; j++) {
    const float ed = ddr[c0 + j];
    denoE += (ed > cmpE) ? ed : 0.f;
  }
  denoE = rsum16(denoE);
  // attn_p = ed*flag / (denoE + 1e-6*sd); rowmask==0 -> all zero
  const float scale_p = rowmask / (rowmask * denoE + 1e-6f * sd);
  const float inv_s0 = 1.0f / s0;

  float* o0 = out_sm0 + ((size_t)bh * N_ + ig) * N_;
  for (int j = 0; j < 64; j++) {
    const int c = c0 + j;
    __builtin_nontemporal_store(d0r[c] * inv_s0, &o0[c]);  // softmax(dots0): 134 MB stream
    const float ed = ddr[c];
    ddr[c] = (ed > cmpE ? ed : 0.f) * scale_p;             // attn_p, A-matrix for PV
  }
  __syncthreads();

  // Phase D: out_head[16,64] = attn_p[16,1024] @ v[1024,64]; K split across wave pairs
  const int ntile = wv & 3, ksel = wv >> 2;
  v8f acc = {0.f, 0.f, 0.f, 0.f, 0.f, 0.f, 0.f, 0.f};
#pragma unroll 4
  for (int kk = 0; kk < 128; kk++) {
    const int k0 = ksel * 512 + kk * 4 + kh * 2;
    v2f a = *(const v2f*)&sm_dd[ln * 1024 + k0];
    v2f bv;
    bv.x = vb[(size_t)k0 * DH_ + ntile * 16 + ln];
    bv.y = vb[(size_t)(k0 + 1) * DH_ + ntile * 16 + ln];
    acc = WMMA_F32X4(a, bv, acc);
  }
  float* sm_part = sm_q;  // sm_q dead after phase C; 4*256 floats fits exactly
  if (wv >= 4) {
#pragma unroll
    for (int r = 0; r < 8; r++) sm_part[ntile * 256 + (r + 8 * kh) * 16 + ln] = acc[r];
  }
  __syncthreads();
  if (wv < 4) {
#pragma unroll
    for (int r = 0; r < 8; r++) {
      const int mr = r + 8 * kh;
      const float val = acc[r] + sm_part[ntile * 256 + mr * 16 + ln];
      out_heads[((size_t)(bb * N_ + m0 + mr)) * (H_ * DH_) + hh * DH_ + ntile * 16 + ln] = val;
    }
  }
}

// ---------------- Kernel 3: output projection ----------------
// out_heads [4096,512] @ w_out [512,512] + b_out
__global__ __launch_bounds__(256) void k_outproj(const float* __restrict__ a,
                                                 const float* __restrict__ w,
                                                 const float* __restrict__ bias,
                                                 float* __restrict__ out) {
  __shared__ float smA[16 * 512];
  const int tid = threadIdx.x, wv = tid >> 5, lane = tid & 31, kh = lane >> 4, ln = lane & 15;
  const int m0 = blockIdx.x * 16;

  if (wv == 0) {
    tdm_load_2d_f32(a + (size_t)m0 * 512, (unsigned)(size_t)(void*)smA,
                    /*tile_w=*/512, /*tile_h=*/16,
                    /*tensor_w=*/512, /*tensor_h=*/4096, /*row_stride=*/512);
    __builtin_amdgcn_s_wait_tensorcnt(0);
  }
  __syncthreads();

  const int col0 = (blockIdx.y * 8 + wv) * 16;
  v8f acc = {0.f, 0.f, 0.f, 0.f, 0.f, 0.f, 0.f, 0.f};
#pragma unroll 4
  for (int kk = 0; kk < 128; kk++) {
    const int k0 = kk * 4 + kh * 2;
    v2f av = *(const v2f*)&smA[ln * 512 + k0];
    v2f bv;
    bv.x = w[(size_t)k0 * 512 + col0 + ln];
    bv.y = w[(size_t)(k0 + 1) * 512 + col0 + ln];
    acc = WMMA_F32X4(av, bv, acc);
  }
  const float bo = bias[col0 + ln];
#pragma unroll
  for (int r = 0; r < 8; r++) {
    const int mr = r + 8 * kh;
    __builtin_nontemporal_store(acc[r] + bo, &out[(size_t)(m0 + mr) * 512 + col0 + ln]);
  }
}

extern "C" void kernel_launch(void* const* d_in, const int* in_sizes, int n_in,
                              void* d_out, int out_size, void* d_ws, size_t ws_size,
                              hipStream_t stream) {
  (void)in_sizes; (void)n_in; (void)out_size; (void)ws_size;
  const float* x      = (const float*)d_in[0];
  const float* prob   = (const float*)d_in[1];
  const float* w_qkv  = (const float*)d_in[2];
  const float* rpb    = (const float*)d_in[3];
  const float* hsita  = (const float*)d_in[4];
  const float* w_th   = (const float*)d_in[5];
  const float* w_out  = (const float*)d_in[6];
  const float* b_out  = (const float*)d_in[7];
  const int*   rpi    = (const int*)d_in[8];
  const float* dis    = (const float*)d_in[9];

  float* out     = (float*)d_out;
  float* out_sm0 = out + (size_t)B_ * N_ * DIM_;  // second tuple output

  float* ws  = (float*)d_ws;                      // 32 MB used
  float* qws = ws;
  float* kws = ws + (size_t)2 * 1024 * 1024;
  float* vws = ws + (size_t)4 * 1024 * 1024;
  float* oh  = ws + (size_t)6 * 1024 * 1024;

  k_qkv<<<dim3(256, 12), 256, 0, stream>>>(x, w_qkv, qws, kws, vws);

  const size_t smem2 = (size_t)(1024 + 2 * 16 * 1024) * sizeof(float);  // 132 KB of 320 KB LDS
  k_attn<<<dim3(2048), 256, smem2, stream>>>(qws, kws, vws, prob, rpb, hsita, w_th, rpi, dis,
                                             oh, out_sm0);

  k_outproj<<<dim3(256, 4), 256, 0, stream>>>(oh, w_out, b_out, out);
}